// FaNatTp_81200651698179
// MI455X (gfx1250) — compile-verified
//
#include <hip/hip_runtime.h>
#include <hip/hip_bf16.h>
#include <stdint.h>

#define LOGZERO (-1.0e10f)
#define MAX_X 512              // xmax capacity
#define MAX_S 160              // state capacity = 32 lanes * 5 states (L = 2*ymax+1 <= 160)
#define NBUF  4                // async gather pipeline depth (read t, issue t+3)

struct SharedState {
    float lpp[NBUF][MAX_S];        // async-DMA-gathered log-probs, 4-deep pipeline
    float alpha[MAX_S];            // final alpha dump (for score/init_state)
    int   path[MAX_S];             // blank-interleaved label path
    unsigned short bp16[MAX_X * 32]; // per (t,lane): 5 x 2-bit argmax codes
    int astate[MAX_X];             // backtraced path-state per time step
    int raw[MAX_X];                // labels along alignment
    int col[MAX_X];                // repeat-collapsed labels
    int trig[MAX_X];               // cumsum of shifted non-blank indicator
    float s_score;
};

__global__ __launch_bounds__(256)
void ctc_viterbi_align_kernel(const float* __restrict__ ctc,
                              const int* __restrict__ src_size,
                              const int* __restrict__ ys,
                              const int* __restrict__ ylens,
                              const int* __restrict__ blank_p,
                              float* __restrict__ out,
                              int bs, int xmax, int vocab, int ymax)
{
    __shared__ SharedState sh;
    const int b   = blockIdx.x;
    const int tid = threadIdx.x;
    const int L   = 2 * ymax + 1;
    if (L > MAX_S || xmax > MAX_X) return;

    const int blank = blank_p[0];
    const int S     = src_size[b];          // DP only needs t in [0, S)
    const int plen  = 2 * ylens[b] + 1;

    // path table for post-processing (read only after the post-DP barrier)
    if (tid < L)
        sh.path[tid] = (tid & 1) ? ys[(size_t)b * ymax + (tid >> 1)] : blank;

    const float* gbase = ctc + (size_t)b * xmax * vocab;

    // =============== single-wave Viterbi DP: lane owns states 5l..5l+4 ===============
    if (tid < 32) {
        const int lane = tid;
        int   pv[5];
        bool  sf[5];
        float a[5];
        #pragma unroll
        for (int j = 0; j < 5; ++j) {
            const int s  = 5 * lane + j;
            const bool odd = (s & 1);
            pv[j] = (odd && s < L) ? ys[(size_t)b * ymax + (s >> 1)] : blank;
            if (!odd)                 sf[j] = true;   // blank->blank: same transition
            else if (s >= 3 && s < L) sf[j] = (pv[j] == ys[(size_t)b * ymax + (s >> 1) - 1]);
            else                      sf[j] = false;
            a[j] = (s == 0) ? 0.0f : LOGZERO;
        }

        // CDNA5 async-DMA gather: lane pulls its 5 states' log-probs straight into LDS.
        // Invalid states (s >= L) fetch the blank column into unused slots: EXEC stays full.
        auto issue = [&](int t) {
            #pragma unroll
            for (int j = 0; j < 5; ++j) {
                unsigned long long gp = (unsigned long long)(const void*)
                    (gbase + (size_t)t * vocab + pv[j]);
                unsigned dst = (unsigned)(uintptr_t)&sh.lpp[t & (NBUF - 1)][5 * lane + j];
                asm volatile("global_load_async_to_lds_b32 %0, %1, off"
                             :: "v"(dst), "v"(gp) : "memory");
            }
        };

        for (int tp = 0; tp < 3 && tp < S; ++tp) issue(tp);   // prime 3-deep pipeline

        for (int t = 0; t < S; ++t) {
            if (t + 3 < S) issue(t + 3);
            // 5 async loads per step retire in order: wait until step t's group landed
            const int remaining = S - 1 - t;
            if (remaining >= 3)      asm volatile("s_wait_asynccnt 0xf" ::: "memory");
            else if (remaining == 2) asm volatile("s_wait_asynccnt 0xa" ::: "memory");
            else if (remaining == 1) asm volatile("s_wait_asynccnt 0x5" ::: "memory");
            else                     asm volatile("s_wait_asynccnt 0x0" ::: "memory");

            float lp[5];
            #pragma unroll
            for (int j = 0; j < 5; ++j) lp[j] = sh.lpp[t & (NBUF - 1)][5 * lane + j];

            // neighbors from previous lane (alpha[5l-1], alpha[5l-2]); lane0 guarded below
            const float p4 = __shfl_up(a[4], 1, 32);
            const float p3 = __shfl_up(a[3], 1, 32);

            float na[5];
            unsigned packed = 0;
            #pragma unroll
            for (int j = 0; j < 5; ++j) {
                const int s = 5 * lane + j;
                float m0 = a[j];
                float m1 = (j >= 1) ? a[j - 1] : p4;
                float m2 = (j >= 2) ? a[j - 2] : ((j == 1) ? p4 : p3);
                if (s < 1)            m1 = LOGZERO;
                if (s < 2 || sf[j])   m2 = LOGZERO;
                float best = m0; int code = 0;       // first-max tie-break == jnp.argmax
                if (m1 > best) { best = m1; code = 1; }
                if (m2 > best) { best = m2; code = 2; }
                const float mp = (s >= plen) ? LOGZERO : best;
                na[j] = mp + lp[j];
                packed |= (unsigned)code << (2 * j);
            }
            #pragma unroll
            for (int j = 0; j < 5; ++j) a[j] = na[j];
            sh.bp16[t * 32 + lane] = (unsigned short)packed;
        }

        #pragma unroll
        for (int j = 0; j < 5; ++j) {
            const int s = 5 * lane + j;
            sh.alpha[s] = (s < L) ? a[j] : LOGZERO;
        }
    }
    __syncthreads();

    // ---- init state, score, sequential backtrace (thread 0) ----
    if (tid == 0) {
        const float s1 = sh.alpha[plen - 1];
        const float s2 = sh.alpha[plen - 2];
        int cur = (s1 > s2) ? (plen - 1) : (plen - 2);
        sh.s_score = fmaxf(s1, s2);
        for (int t = S - 1; t >= 0; --t) {
            sh.astate[t] = cur;
            if (t >= 1) {
                const unsigned v = sh.bp16[t * 32 + (cur / 5)];
                cur -= (int)((v >> (2 * (cur % 5))) & 3u);
            }
        }
    }
    for (int t = S + tid; t < xmax; t += blockDim.x) sh.astate[t] = 0;
    __syncthreads();

    // ---- map to labels, collapse repeats ----
    for (int t = tid; t < xmax; t += blockDim.x)
        sh.raw[t] = sh.path[sh.astate[t]];
    __syncthreads();
    for (int t = tid; t < xmax; t += blockDim.x) {
        const int prev = (t == 0) ? 0 : sh.raw[t - 1];
        sh.col[t] = (sh.raw[t] == prev) ? 0 : sh.raw[t];
    }
    __syncthreads();

    // ---- trig[t] = inclusive cumsum of shifted non-blank indicator ----
    if (tid == 0) {
        int acc = 0;
        for (int t = 0; t < xmax; ++t) {
            const int shifted = (t == 0) ? 0 : sh.col[t - 1];
            acc += (shifted != blank) ? 1 : 0;
            sh.trig[t] = acc;
        }
    }
    __syncthreads();

    // ---- outputs (flat float, in return order) ----
    float* out_seq   = out;
    float* out_trig  = out + (size_t)bs * xmax;
    float* out_ylen  = out_trig + (size_t)bs * (ymax + 1) * xmax;
    float* out_score = out_ylen + bs;

    for (int t = tid; t < xmax; t += blockDim.x)
        out_seq[(size_t)b * xmax + t] = (float)sh.col[t];

    const int tot = (ymax + 1) * xmax;
    for (int i = tid; i < tot; i += blockDim.x) {
        const int y = i / xmax;
        const int t = i - y * xmax;
        bool v;
        if (y == ymax)
            v = ((sh.trig[t] == y) && (t < S)) || (t == S - 1);
        else
            v = (sh.trig[t] == y);
        out_trig[((size_t)b * (ymax + 1) + y) * xmax + t] = v ? 1.0f : 0.0f;
    }

    if (tid == 0) {
        out_ylen[b]  = (float)(ylens[b] + 1);
        out_score[b] = sh.s_score;
    }
}

extern "C" void kernel_launch(void* const* d_in, const int* in_sizes, int n_in,
                              void* d_out, int out_size, void* d_ws, size_t ws_size,
                              hipStream_t stream) {
    (void)n_in; (void)out_size; (void)d_ws; (void)ws_size;
    const float* ctc      = (const float*)d_in[0];
    // d_in[1] = src_mask (bool) — redundant with src_size, unused
    const int*   src_size = (const int*)d_in[2];
    const int*   ys       = (const int*)d_in[3];
    const int*   ylens    = (const int*)d_in[4];
    const int*   blank    = (const int*)d_in[5];

    const int bs    = in_sizes[2];
    const int ymax  = in_sizes[3] / bs;
    const int xmax  = in_sizes[1] / bs;              // src_mask is [bs,1,xmax]
    const int vocab = in_sizes[0] / (bs * xmax);

    dim3 grid(bs), block(256);
    ctc_viterbi_align_kernel<<<grid, block, 0, stream>>>(
        ctc, src_size, ys, ylens, blank, (float*)d_out,
        bs, xmax, vocab, ymax);
}